// VectorQuantizedPSI_Audio_1941325218350
// MI455X (gfx1250) — compile-verified
//
#include <hip/hip_runtime.h>
#include <hip/hip_bf16.h>

// ---------------------------------------------------------------------------
// VQ-VAE forward (VectorQuantizedPSI_Audio) for gfx1250 / MI455X.
// fp32 WMMA (V_WMMA_F32_16X16X4_F32) for the VQ distance GEMM and the four
// 128->128 ConvTranspose layers (parity-decomposed implicit GEMM).
// GLOBAL_LOAD_ASYNC_TO_LDS staging + s_wait_asynccnt, ds_min_u64 argmin,
// per-layer weight transpose for b64 B-fragment loads.
// ---------------------------------------------------------------------------

typedef float v2f __attribute__((ext_vector_type(2)));
typedef float v8f __attribute__((ext_vector_type(8)));

static __device__ __forceinline__ v8f wmma_f32_k4(v2f a, v2f b, v8f c) {
  // D = A(16x4 f32) * B(4x16 f32) + C(16x16 f32)
  return __builtin_amdgcn_wmma_f32_16x16x4_f32(false, a, false, b, (short)0, c,
                                               false, false);
}

// ----------------------- problem constants ---------------------------------
#define BB 4
#define CC 128
#define HH 32
#define WW 32
#define KK 512
#define NSHARED 5
#define BN_EPS 1e-5f

// linspace step in f32, matching jnp.linspace(-0.5, 49.49, 507)
#define VQ_STEP (49.99f / 506.0f)

// layer spatial sizes
#define H1 63
#define H2 126
#define H3 252
#define H4 504
#define H5 513

// workspace layout (float offsets)
#define WS_ZQ 0ull                          // 524288
#define WS_CBN 524288ull                    // 512
#define WS_IDX 524800ull                    // 4096 (int)
#define WS_STATS 528896ull                  // 256
#define WS_WT 529152ull                     // 262144 (transposed weights)
#define WS_B1 791296ull                     // 2032128
#define WS_B2 2823424ull                    // 8128512
#define WS_B3 10951936ull                   // 32514048
#define WS_B4 43465984ull                   // 130056192

// output layout (float offsets)
#define OUT_XT 0ull                         // 4*1*513*513 = 1052676
#define OUT_HCAT 1052676ull                 // 524288
#define OUT_ZQ 1576964ull                   // 524288

// ---------------------------------------------------------------------------
// small utility kernels
// ---------------------------------------------------------------------------
__global__ __launch_bounds__(256) void zero_f32(float* p, int n) {
  int i = blockIdx.x * 256 + threadIdx.x;
  if (i < n) p[i] = 0.0f;
}

__global__ __launch_bounds__(256) void copy_f32(const float* __restrict__ a,
                                                float* __restrict__ b, int n) {
  int i = blockIdx.x * 256 + threadIdx.x;
  if (i < n) b[i] = a[i];
}

// per-code squared norms of the codebook
__global__ __launch_bounds__(256) void cb_norm_kernel(const float* __restrict__ cb,
                                                      float* __restrict__ cbn) {
  int k = blockIdx.x * 256 + threadIdx.x;
  if (k >= KK) return;
  float s = 0.0f;
  const float* row = cb + (size_t)k * CC;
#pragma unroll 8
  for (int c = 0; c < CC; ++c) s = fmaf(row[c], row[c], s);
  cbn[k] = s;
}

// transpose dec_w [Cin][Cout][KS][KS] -> wT [tap][Cout][Cin] (b64-friendly B)
__global__ __launch_bounds__(256) void wt_transpose_kernel(const float* __restrict__ w,
                                                           float* __restrict__ wT,
                                                           int KS2) {
  int i = blockIdx.x * 256 + threadIdx.x;
  int n = KS2 * CC * CC;
  if (i >= n) return;
  int tap = i / (CC * CC);
  int r = i - tap * (CC * CC);
  int co = r >> 7;
  int c = r & 127;
  wT[i] = w[(size_t)(c * CC + co) * KS2 + tap];
}

// ---------------------------------------------------------------------------
// VQ: WMMA distance GEMM + masked argmin.
// One workgroup = 16 pixels (one image => one label). 8 waves x 4 code-tiles.
// ---------------------------------------------------------------------------
__global__ __launch_bounds__(256) void vq_kernel(const float* __restrict__ hs,
                                                 const float* __restrict__ cb,
                                                 const int* __restrict__ labels,
                                                 const float* __restrict__ cbn,
                                                 int* __restrict__ idx_out) {
  __shared__ float zs[CC][16];                 // [channel][row-in-tile]
  __shared__ unsigned long long best[16];

  const int tid = threadIdx.x;
  const int row0 = blockIdx.x * 16;
  const int b = row0 >> 10;
  const int hw0 = row0 & 1023;
  const float lab = (float)labels[b];

  for (int e = tid; e < CC * 16; e += 256) {
    int m = e & 15, c = e >> 4;
    zs[c][m] = hs[((size_t)(b * CC + c) << 10) + hw0 + m];
  }
  if (tid < 16) best[tid] = ~0ull;
  __syncthreads();

  const int wv = tid >> 5;
  const int lane = tid & 31;
  const int hf = lane >> 4;
  const int nm = lane & 15;

  unsigned long long loc[8];
#pragma unroll
  for (int i = 0; i < 8; ++i) loc[i] = ~0ull;

#pragma unroll
  for (int t = 0; t < 4; ++t) {
    const int kt = wv * 4 + t;
    const int code = kt * 16 + nm;
    v8f acc = {0.f, 0.f, 0.f, 0.f, 0.f, 0.f, 0.f, 0.f};
    const float* crow = cb + (size_t)code * CC;
#pragma unroll 4
    for (int s = 0; s < 32; ++s) {
      const int c0 = 4 * s + 2 * hf;
      v2f a, bf;
      a.x = zs[c0][nm];
      a.y = zs[c0 + 1][nm];
      bf.x = crow[c0];
      bf.y = crow[c0 + 1];
      acc = wmma_f32_k4(a, bf, acc);
    }
    // class-partition mask (shared codes are masked in TRAIN stage)
    bool allowed = false;
    if (code < KK - NSHARED) {
      float u = __fadd_rn(__fmul_rn((float)code, VQ_STEP), -0.5f);
      allowed = (nearbyintf(u) == lab);
    }
    if (allowed) {
      const float cn = cbn[code];
#pragma unroll
      for (int i = 0; i < 8; ++i) {
        float score = fmaf(-2.0f, acc[i], cn);   // |z|^2 dropped (row const)
        unsigned u = __float_as_uint(score);
        u = (u & 0x80000000u) ? ~u : (u | 0x80000000u);
        unsigned long long key = ((unsigned long long)u << 32) | (unsigned)code;
        loc[i] = loc[i] < key ? loc[i] : key;
      }
    }
  }
#pragma unroll
  for (int i = 0; i < 8; ++i) atomicMin(&best[i + 8 * hf], loc[i]);  // ds_min_u64
  __syncthreads();
  if (tid < 16) idx_out[row0 + tid] = (int)(best[tid] & 0xFFFFFFFFull);
}

// gather codes -> NCHW z_q (workspace copy for the decoder + d_out)
__global__ __launch_bounds__(256) void gather_kernel(const float* __restrict__ cb,
                                                     const int* __restrict__ idx,
                                                     float* __restrict__ zq_ws,
                                                     float* __restrict__ zq_out) {
  int i = blockIdx.x * 256 + threadIdx.x;
  if (i >= BB * CC * HH * WW) return;
  int hw = i & 1023;
  int c = (i >> 10) & 127;
  int b = i >> 17;
  int n = (b << 10) + hw;
  float v = cb[(size_t)idx[n] * CC + c];
  zq_ws[i] = v;
  zq_out[i] = v;
}

// ---------------------------------------------------------------------------
// ConvTranspose2d (stride 2, pad 1, Cin=Cout=128), parity-decomposed WMMA
// implicit GEMM. Workgroup: (b, oy, parity px, 16 same-parity ox). 8 waves =
// 8 N-tiles of 16 channels. Activations staged via GLOBAL_LOAD_ASYNC_TO_LDS.
// Fused bias + ReLU epilogue.
// ---------------------------------------------------------------------------
template <int KS>
__global__ __launch_bounds__(256) void convt_wmma(const float* __restrict__ x,
                                                  const float* __restrict__ wT,
                                                  const float* __restrict__ bias,
                                                  float* __restrict__ y,
                                                  int Hin, int Win,
                                                  int Hout, int Wout) {
  constexpr int COLS = 18;                     // 16 + halo(1 each side)
  __shared__ float xs[2 * CC * COLS];          // [slot][c][col]

  const int tid = threadIdx.x;
  const int px = blockIdx.x & 1;               // output column parity
  const int tq = blockIdx.x >> 1;              // same-parity 16-col tile
  const int oy = blockIdx.y;
  const int b = blockIdx.z;

  // valid input rows for this output row (<=2 due to stride-2 parity)
  int slot_of_ky[KS];
  int iy0 = 0, iy1 = 0, nslots = 0;
#pragma unroll
  for (int ky = 0; ky < KS; ++ky) {
    int t = oy + 1 - ky;
    if (t >= 0 && (t & 1) == 0 && (t >> 1) < Hin) {
      slot_of_ky[ky] = nslots;
      if (nslots == 0) iy0 = t >> 1; else iy1 = t >> 1;
      ++nslots;
    } else {
      slot_of_ky[ky] = -1;
    }
  }
  const int ixbase = tq * 16 - 1;

  // async stage of input rows into LDS (zero-fill the out-of-range halo)
  const int total = nslots * CC * COLS;
  for (int e = tid; e < total; e += 256) {
    int slot = e / (CC * COLS);
    int r = e - slot * (CC * COLS);
    int c = r / COLS;
    int col = r - c * COLS;
    int ix = ixbase + col;
    int iy = slot ? iy1 : iy0;
    float* dst = &xs[slot * CC * COLS + c * COLS + col];
    if ((unsigned)ix < (unsigned)Win) {
      const float* src = &x[((size_t)(b * CC + c) * Hin + iy) * Win + ix];
      unsigned ldsa = (unsigned)(uintptr_t)dst;          // low 32b = LDS offset
      unsigned long long ga = (unsigned long long)(uintptr_t)src;
      asm volatile("global_load_async_to_lds_b32 %0, %1, off"
                   :: "v"(ldsa), "v"(ga) : "memory");
    } else {
      *dst = 0.0f;
    }
  }
  asm volatile("s_wait_asynccnt 0x0" ::: "memory");
  __syncthreads();

  const int wv = tid >> 5;
  const int lane = tid & 31;
  const int hf = lane >> 4;
  const int nm = lane & 15;
  const int co = wv * 16 + nm;

  v8f acc = {0.f, 0.f, 0.f, 0.f, 0.f, 0.f, 0.f, 0.f};

#pragma unroll
  for (int ky = 0; ky < KS; ++ky) {
    const int slot = slot_of_ky[ky];
    if (slot < 0) continue;
    const float* xb = &xs[slot * CC * COLS];
#pragma unroll
    for (int kx = 0; kx < KS; ++kx) {
      if (((px + 1 - kx) & 1) != 0) continue;  // wrong parity tap (uniform)
      const int d = (px + 1 - kx) >> 1;        // in {-1,0,1}
      const int ixm = tq * 16 + nm + d;        // this M-row's input column
      const bool mval = (unsigned)ixm < (unsigned)Win;
      const int col = nm + d + 1;
      const float* wt = wT + (size_t)((ky * KS + kx) * CC + co) * CC;
#pragma unroll 4
      for (int s = 0; s < 32; ++s) {
        const int c0 = 4 * s + 2 * hf;
        v2f a, bf;
        a.x = mval ? xb[c0 * COLS + col] : 0.0f;
        a.y = mval ? xb[(c0 + 1) * COLS + col] : 0.0f;
        bf.x = wt[c0];                          // contiguous pair -> b64
        bf.y = wt[c0 + 1];
        acc = wmma_f32_k4(a, bf, acc);
      }
    }
  }

  // epilogue: bias + ReLU; lane owns 8 same-parity pixels of channel `co`
  const float bv = bias[co];
  const size_t ybase = ((size_t)(b * CC + co) * Hout + oy) * Wout;
  const int m0 = 8 * hf;
#pragma unroll
  for (int i = 0; i < 8; ++i) {
    int ox = 2 * (tq * 16 + m0 + i) + px;
    if (ox < Wout) {
      float v = acc[i] + bv;
      y[ybase + ox] = v > 0.0f ? v : 0.0f;
    }
  }
}

// ---------------------------------------------------------------------------
// BatchNorm (training stats) — per-channel sum/sumsq reduction + normalize
// ---------------------------------------------------------------------------
#define BN_SLICES 16
__global__ __launch_bounds__(256) void bn_stats_kernel(const float* __restrict__ y,
                                                       float* __restrict__ stats,
                                                       int HW) {
  const int c = blockIdx.x & (CC - 1);
  const int slice = blockIdx.x >> 7;
  const int total = BB * HW;
  float s0 = 0.0f, s1 = 0.0f;
  for (int e = slice * 256 + threadIdx.x; e < total; e += BN_SLICES * 256) {
    int b = e / HW;
    int hw = e - b * HW;
    float v = y[((size_t)(b * CC + c)) * HW + hw];
    s0 += v;
    s1 = fmaf(v, v, s1);
  }
  __shared__ float sh0, sh1;
  if (threadIdx.x == 0) { sh0 = 0.0f; sh1 = 0.0f; }
  __syncthreads();
  atomicAdd(&sh0, s0);
  atomicAdd(&sh1, s1);
  __syncthreads();
  if (threadIdx.x == 0) {
    atomicAdd(&stats[2 * c], sh0);
    atomicAdd(&stats[2 * c + 1], sh1);
  }
}

__global__ __launch_bounds__(256) void bn_apply_kernel(float* __restrict__ y,
                                                       const float* __restrict__ stats,
                                                       const float* __restrict__ g,
                                                       const float* __restrict__ be,
                                                       int HW, long long totaln) {
  long long i = (long long)blockIdx.x * 256 + threadIdx.x;
  if (i >= totaln) return;
  int c = (int)((i / HW) & (CC - 1));
  float n = (float)(BB * HW);
  float mean = stats[2 * c] / n;
  float var = stats[2 * c + 1] / n - mean * mean;   // biased, matches jnp.var
  float sc = rsqrtf(var + BN_EPS) * g[c];
  y[i] = (y[i] - mean) * sc + be[c];
}

// ---------------------------------------------------------------------------
// Final ConvTranspose2d: 128 -> 1, k=12, s=1, p=1 : 504x504 -> 513x513.
// Cout=1 kills WMMA N-efficiency; coalesced VALU kernel, L2-resident input.
// ---------------------------------------------------------------------------
__global__ __launch_bounds__(256) void convt_final_kernel(const float* __restrict__ x,
                                                          const float* __restrict__ w,
                                                          const float* __restrict__ bias,
                                                          float* __restrict__ out) {
  const int HWo = H5 * H5;
  int i = blockIdx.x * 256 + threadIdx.x;
  if (i >= BB * HWo) return;
  int b = i / HWo;
  int r = i - b * HWo;
  int oy = r / H5;
  int ox = r - oy * H5;
  float sum = bias[0];
  const size_t plane = (size_t)H4 * H4;
  const float* xb = x + (size_t)b * CC * plane;
#pragma unroll 2
  for (int ky = 0; ky < 12; ++ky) {
    int iy = oy + 1 - ky;
    if ((unsigned)iy >= (unsigned)H4) continue;
#pragma unroll 2
    for (int kx = 0; kx < 12; ++kx) {
      int ix = ox + 1 - kx;
      if ((unsigned)ix >= (unsigned)H4) continue;
      const float* xp = xb + (size_t)iy * H4 + ix;
      const float* wp = w + ky * 12 + kx;
      float a = 0.0f;
#pragma unroll 8
      for (int ci = 0; ci < CC; ++ci)
        a = fmaf(xp[ci * plane], wp[ci * 144], a);
      sum += a;
    }
  }
  out[i] = sum;
}

// ---------------------------------------------------------------------------
// launch
// ---------------------------------------------------------------------------
extern "C" void kernel_launch(void* const* d_in, const int* in_sizes, int n_in,
                              void* d_out, int out_size, void* d_ws, size_t ws_size,
                              hipStream_t stream) {
  const float* hs = (const float*)d_in[0];
  const int* labels = (const int*)d_in[1];
  const float* cb = (const float*)d_in[2];
  const float* dw[5] = {(const float*)d_in[3], (const float*)d_in[4],
                        (const float*)d_in[5], (const float*)d_in[6],
                        (const float*)d_in[7]};
  const float* db[5] = {(const float*)d_in[8], (const float*)d_in[9],
                        (const float*)d_in[10], (const float*)d_in[11],
                        (const float*)d_in[12]};
  const float* bg[4] = {(const float*)d_in[13], (const float*)d_in[14],
                        (const float*)d_in[15], (const float*)d_in[16]};
  const float* bb[4] = {(const float*)d_in[17], (const float*)d_in[18],
                        (const float*)d_in[19], (const float*)d_in[20]};

  float* out = (float*)d_out;
  float* W = (float*)d_ws;

  // --- VQ front-end --------------------------------------------------------
  cb_norm_kernel<<<2, 256, 0, stream>>>(cb, W + WS_CBN);
  vq_kernel<<<(BB * HH * WW) / 16, 256, 0, stream>>>(hs, cb, labels, W + WS_CBN,
                                                     (int*)(W + WS_IDX));
  gather_kernel<<<(BB * CC * HH * WW + 255) / 256, 256, 0, stream>>>(
      cb, (const int*)(W + WS_IDX), W + WS_ZQ, out + OUT_ZQ);
  copy_f32<<<(BB * CC * HH * WW + 255) / 256, 256, 0, stream>>>(
      hs, out + OUT_HCAT, BB * CC * HH * WW);

  struct Layer {
    int ks, Hin, Hout;
    const float* in;
    float* outb;
  };
  Layer L[4] = {{3, HH, H1, W + WS_ZQ, W + WS_B1},
                {4, H1, H2, W + WS_B1, W + WS_B2},
                {4, H2, H3, W + WS_B2, W + WS_B3},
                {4, H3, H4, W + WS_B3, W + WS_B4}};

  for (int li = 0; li < 4; ++li) {
    const Layer& l = L[li];
    int KS2 = l.ks * l.ks;
    wt_transpose_kernel<<<(KS2 * CC * CC + 255) / 256, 256, 0, stream>>>(
        dw[li], W + WS_WT, KS2);
    zero_f32<<<1, 256, 0, stream>>>(W + WS_STATS, 256);
    int halfW = (l.Hout + 1) / 2;
    dim3 grid(2 * ((halfW + 15) / 16), l.Hout, BB);
    if (l.ks == 3)
      convt_wmma<3><<<grid, 256, 0, stream>>>(l.in, W + WS_WT, db[li], l.outb,
                                              l.Hin, l.Hin, l.Hout, l.Hout);
    else
      convt_wmma<4><<<grid, 256, 0, stream>>>(l.in, W + WS_WT, db[li], l.outb,
                                              l.Hin, l.Hin, l.Hout, l.Hout);
    int HW = l.Hout * l.Hout;
    long long totaln = (long long)BB * CC * HW;
    bn_stats_kernel<<<CC * BN_SLICES, 256, 0, stream>>>(l.outb, W + WS_STATS, HW);
    bn_apply_kernel<<<(unsigned)((totaln + 255) / 256), 256, 0, stream>>>(
        l.outb, W + WS_STATS, bg[li], bb[li], HW, totaln);
  }

  // --- final 128 -> 1 layer ------------------------------------------------
  convt_final_kernel<<<(BB * H5 * H5 + 255) / 256, 256, 0, stream>>>(
      W + WS_B4, dw[4], db[4], out + OUT_XT);
}